// BiGCN_A_53996328845503
// MI455X (gfx1250) — compile-verified
//
#include <hip/hip_runtime.h>
#include <cstdint>
#include <cstddef>

#define INC  128
#define HIDC 64
#define OUTC 40

typedef float v2f __attribute__((ext_vector_type(2)));
typedef float v8f __attribute__((ext_vector_type(8)));

// ---------------- degree / norm kernels ----------------

__global__ void k_init_deg(float* __restrict__ deg, int n) {
  int i = blockIdx.x * blockDim.x + threadIdx.x;
  if (i < n) deg[i] = 1.0f;                       // self-loop contribution
}

__global__ void k_count_deg(const int* __restrict__ dst, float* __restrict__ deg, int E) {
  int e = blockIdx.x * blockDim.x + threadIdx.x;
  if (e < E) unsafeAtomicAdd(&deg[dst[e]], 1.0f);
}

__global__ void k_rsqrt(float* __restrict__ d, int n) {
  int i = blockIdx.x * blockDim.x + threadIdx.x;
  if (i < n) d[i] = rsqrtf(d[i]);                 // deg >= 1 always (self-loop)
}

__global__ void k_norm(const int* __restrict__ src, const int* __restrict__ dst,
                       const float* __restrict__ dinv, float* __restrict__ norm, int E) {
  int e = blockIdx.x * blockDim.x + threadIdx.x;
  if (e < E) norm[e] = dinv[src[e]] * dinv[dst[e]];
}

// ---------------- GEMM1: hx[N,64] = x[N,128] @ W1[128,64]  (fp32 WMMA) ----------------
// grid.x = N/16, block = 128 threads (4 waves); wave w owns output cols [16w,16w+16)

__global__ __launch_bounds__(128)
void k_gemm1_wmma(const float* __restrict__ x, const float* __restrict__ W1,
                  float* __restrict__ hx) {
  __shared__ float sA[16 * 132];                  // 16 x 128, pitch 132 (bank-conflict free)
  const int r0 = blockIdx.x * 16;
  const int t  = threadIdx.x;

  // cooperative A-tile load: 2048 floats = 512 float4; 128 threads x 4 float4
  {
    const int row = t >> 3;                       // 0..15
    const int q   = (t & 7) * 16;                 // starting float col, multiple of 16
    const float4* s = (const float4*)(x + (size_t)(r0 + row) * INC + q);
    float4*       d = (float4*)(sA + row * 132 + q);
#pragma unroll
    for (int j = 0; j < 4; ++j) d[j] = s[j];
  }
  __syncthreads();

  const int wave  = t >> 5;                       // 0..3 -> col tile
  const int lane  = t & 31;
  const int c0    = wave * 16;
  const int lhalf = lane & 15;                    // M (A) / N (B,C)
  const int khi   = (lane >> 4) * 2;              // K sub-offset: 0 or 2

  v8f acc = {};
#pragma unroll
  for (int k = 0; k < INC; k += 4) {
    v2f a, b;
    const float* ap = sA + lhalf * 132 + k + khi; // A[M=lhalf][K=k+khi, k+khi+1]
    a.x = ap[0];
    a.y = ap[1];
    const float* bp = W1 + (size_t)(k + khi) * HIDC + c0 + lhalf;
    b.x = bp[0];                                  // B[K=k+khi  ][N=c0+lhalf]
    b.y = bp[HIDC];                               // B[K=k+khi+1][N=c0+lhalf]
    acc = __builtin_amdgcn_wmma_f32_16x16x4_f32(false, a, false, b,
                                                (short)0, acc, false, false);
  }

  // C layout: VGPR v, lanes 0-15 -> M=v; lanes 16-31 -> M=v+8; N = lhalf
  const int m0 = r0 + ((lane >> 4) * 8);
  const int n  = c0 + lhalf;
#pragma unroll
  for (int v = 0; v < 8; ++v)
    hx[(size_t)(m0 + v) * HIDC + n] = acc[v];
}

// ---------------- GEMM2: h2[N,40] = relu(agg1 + b1)[N,64] @ W2[64,40] ----------------
// grid.x = N/16, block = 96 threads (3 waves); tile cols 0..47, predicated at col>=40

__global__ __launch_bounds__(96)
void k_gemm2_wmma(const float* __restrict__ agg1, const float* __restrict__ b1,
                  const float* __restrict__ W2, float* __restrict__ h2) {
  __shared__ float sA[16 * 68];                   // 16 x 64, pitch 68
  const int r0 = blockIdx.x * 16;
  const int t  = threadIdx.x;

  // stage A-tile with fused bias + ReLU: 1024 floats = 256 float4
  for (int i = t; i < 256; i += 96) {
    const int row = i >> 4;                       // 16 float4 per row
    const int q   = (i & 15) * 4;
    float4 v = *(const float4*)(agg1 + (size_t)(r0 + row) * HIDC + q);
    v.x = fmaxf(v.x + b1[q + 0], 0.0f);
    v.y = fmaxf(v.y + b1[q + 1], 0.0f);
    v.z = fmaxf(v.z + b1[q + 2], 0.0f);
    v.w = fmaxf(v.w + b1[q + 3], 0.0f);
    *(float4*)(sA + row * 68 + q) = v;
  }
  __syncthreads();

  const int wave  = t >> 5;
  const int lane  = t & 31;
  const int c0    = wave * 16;
  const int lhalf = lane & 15;
  const int khi   = (lane >> 4) * 2;
  const int col   = c0 + lhalf;
  const bool colOK = (col < OUTC);

  v8f acc = {};
#pragma unroll
  for (int k = 0; k < HIDC; k += 4) {
    v2f a, b;
    const float* ap = sA + lhalf * 68 + k + khi;
    a.x = ap[0];
    a.y = ap[1];
    float bx = 0.0f, by = 0.0f;
    if (colOK) {                                  // guard only the loads; WMMA sees full EXEC
      bx = W2[(size_t)(k + khi) * OUTC + col];
      by = W2[(size_t)(k + khi + 1) * OUTC + col];
    }
    b.x = bx;
    b.y = by;
    acc = __builtin_amdgcn_wmma_f32_16x16x4_f32(false, a, false, b,
                                                (short)0, acc, false, false);
  }

  if (colOK) {
    const int m0 = r0 + ((lane >> 4) * 8);
#pragma unroll
    for (int v = 0; v < 8; ++v)
      h2[(size_t)(m0 + v) * OUTC + col] = acc[v];
  }
}

// ---------------- aggregation (self-loop init + edge scatter-add) ----------------
// All vectorized: 1 thread = one 4-channel (float4) chunk; atomics stay scalar f32
// (no packed f32 atomic exists) but index/norm loads are amortized 4x.

__global__ void k_selfloop_h(const float* __restrict__ hx, const float* __restrict__ dinv,
                             float* __restrict__ agg, int total4) { // total4 = N*16
  int t = blockIdx.x * blockDim.x + threadIdx.x;
  if (t < total4) {
    int i = t >> 4;                                // node
    float di = dinv[i];
    float s = di * di;
    float4 v = ((const float4*)hx)[t];
    v.x *= s; v.y *= s; v.z *= s; v.w *= s;
    ((float4*)agg)[t] = v;
  }
}

__global__ void k_scatter_h(const int* __restrict__ src, const int* __restrict__ dst,
                            const float* __restrict__ norm, const float* __restrict__ hx,
                            float* __restrict__ agg, int total4) { // total4 = E*16
  int t = blockIdx.x * blockDim.x + threadIdx.x;
  if (t < total4) {
    int e  = t >> 4;
    int c4 = (t & 15) * 4;
    float w = norm[e];
    const float4 v = *(const float4*)(hx + (size_t)src[e] * HIDC + c4);
    float* p = agg + (size_t)dst[e] * HIDC + c4;
    unsafeAtomicAdd(p + 0, v.x * w);
    unsafeAtomicAdd(p + 1, v.y * w);
    unsafeAtomicAdd(p + 2, v.z * w);
    unsafeAtomicAdd(p + 3, v.w * w);
  }
}

__global__ void k_selfloop_o(const float* __restrict__ h2, const float* __restrict__ dinv,
                             const float* __restrict__ b2, float* __restrict__ out,
                             int total4) {                          // total4 = N*10
  int t = blockIdx.x * blockDim.x + threadIdx.x;
  if (t < total4) {
    int i = t / 10;                                // node
    int q = (t - i * 10) * 4;                      // channel chunk
    float di = dinv[i];
    float s = di * di;
    float4 v = ((const float4*)h2)[t];             // h2 rows are 40 floats = 10 float4
    const float4 b = *(const float4*)(b2 + q);
    v.x = b.x + s * v.x;
    v.y = b.y + s * v.y;
    v.z = b.z + s * v.z;
    v.w = b.w + s * v.w;
    ((float4*)out)[t] = v;
  }
}

__global__ void k_scatter_o(const int* __restrict__ src, const int* __restrict__ dst,
                            const float* __restrict__ norm, const float* __restrict__ h2,
                            float* __restrict__ out, int total4) {  // total4 = E*10
  int t = blockIdx.x * blockDim.x + threadIdx.x;
  if (t < total4) {
    int e = t / 10;
    int q = (t - e * 10) * 4;
    float w = norm[e];
    const float4 v = *(const float4*)(h2 + (size_t)src[e] * OUTC + q);
    float* p = out + (size_t)dst[e] * OUTC + q;
    unsafeAtomicAdd(p + 0, v.x * w);
    unsafeAtomicAdd(p + 1, v.y * w);
    unsafeAtomicAdd(p + 2, v.z * w);
    unsafeAtomicAdd(p + 3, v.w * w);
  }
}

// ---------------- host launcher ----------------

extern "C" void kernel_launch(void* const* d_in, const int* in_sizes, int n_in,
                              void* d_out, int out_size, void* d_ws, size_t ws_size,
                              hipStream_t stream) {
  const float* x  = (const float*)d_in[0];
  const int*   ei = (const int*)d_in[1];          // [2, E] flat
  const float* W1 = (const float*)d_in[2];
  const float* b1 = (const float*)d_in[3];
  const float* W2 = (const float*)d_in[4];
  const float* b2 = (const float*)d_in[5];
  float* out = (float*)d_out;

  const int N = in_sizes[0] / INC;                // 100000 (multiple of 16)
  const int E = in_sizes[1] / 2;                  // 1600000
  const int* src = ei;
  const int* dst = ei + E;

  // workspace carve-out (~74.0 MB)
  char*  ws  = (char*)d_ws;
  size_t off = 0;
  auto carve = [&](size_t bytes) -> float* {
    float* p = (float*)(ws + off);
    off = (off + bytes + 255) & ~(size_t)255;
    return p;
  };
  float* dinv = carve((size_t)N * 4);             // deg -> rsqrt(deg), in place
  float* norm = carve((size_t)E * 4);
  float* hx   = carve((size_t)N * HIDC * 4);
  float* agg1 = carve((size_t)N * HIDC * 4);
  float* h2   = carve((size_t)N * OUTC * 4);
  (void)n_in; (void)out_size; (void)ws_size;

  const int B = 256;
  // 1) degrees (self-loop folded into init) and symmetric norm
  k_init_deg  <<<(N + B - 1) / B, B, 0, stream>>>(dinv, N);
  k_count_deg <<<(E + B - 1) / B, B, 0, stream>>>(dst, dinv, E);
  k_rsqrt     <<<(N + B - 1) / B, B, 0, stream>>>(dinv, N);
  k_norm      <<<(E + B - 1) / B, B, 0, stream>>>(src, dst, dinv, norm, E);

  // 2) layer 1: GEMM (WMMA f32) -> self-loop init -> edge scatter-add
  k_gemm1_wmma<<<N / 16, 128, 0, stream>>>(x, W1, hx);
  {
    int tot4 = N * (HIDC / 4);                    // 1.6M float4 chunks
    k_selfloop_h<<<(tot4 + B - 1) / B, B, 0, stream>>>(hx, dinv, agg1, tot4);
  }
  {
    int tot4 = E * (HIDC / 4);                    // 25.6M float4 chunks
    k_scatter_h<<<(tot4 + B - 1) / B, B, 0, stream>>>(src, dst, norm, hx, agg1, tot4);
  }

  // 3) layer 2: fused (bias+ReLU) GEMM (WMMA f32) -> self-loop+bias init -> scatter-add
  k_gemm2_wmma<<<N / 16, 96, 0, stream>>>(agg1, b1, W2, h2);
  {
    int tot4 = N * (OUTC / 4);                    // 1.0M float4 chunks
    k_selfloop_o<<<(tot4 + B - 1) / B, B, 0, stream>>>(h2, dinv, b2, out, tot4);
  }
  {
    int tot4 = E * (OUTC / 4);                    // 16M float4 chunks
    k_scatter_o<<<(tot4 + B - 1) / B, B, 0, stream>>>(src, dst, norm, h2, out, tot4);
  }
}